// GCNEncoder2_35201551958715
// MI455X (gfx1250) — compile-verified
//
#include <hip/hip_runtime.h>

typedef __attribute__((ext_vector_type(2))) float v2f;
typedef __attribute__((ext_vector_type(8))) float v8f;

// ---------------------------------------------------------------------------
// Degree: deg[dst] += 1  (float adds of small ints are exact)
// ---------------------------------------------------------------------------
__global__ __launch_bounds__(256)
void degree_kernel(const int* __restrict__ dst, float* __restrict__ deg, int E) {
    long t = (long)blockIdx.x * blockDim.x + threadIdx.x;
    if (t < E) unsafeAtomicAdd(&deg[dst[t]], 1.0f);
}

// dinv[i] = rsqrt(deg[i] + 1)   (in place over deg buffer)
__global__ __launch_bounds__(256)
void rsqrt_kernel(float* __restrict__ d, int n) {
    long t = (long)blockIdx.x * blockDim.x + threadIdx.x;
    if (t < n) d[t] = rsqrtf(d[t] + 1.0f);
}

// ---------------------------------------------------------------------------
// FP32 WMMA GEMM:  C[M,NOUT] = A[M,K] * W[K,NOUT]
// One 16x16 C tile per wave, V_WMMA_F32_16X16X4_F32, K-loop step 4.
// A-frag (16x4, 32-bit): lane&15 = M row; lane>>4 picks K pair {0,1}/{2,3};
//   vgpr0/1 = first/second K of the pair (contiguous in row-major A -> float2).
// B-frag (4x16): lane&15 = N col; lane>>4 picks K pair; vgpr0/1 = K within pair.
// C/D (16x16, 32-bit): vgpr g -> row g + 8*(lane>>4), col = lane&15.
// ---------------------------------------------------------------------------
template<int K, int NOUT>
__global__ __launch_bounds__(256)
void gemm_wmma_f32(const float* __restrict__ A, const float* __restrict__ W,
                   float* __restrict__ C, int M) {
    const int wave = threadIdx.x >> 5;
    const int lane = threadIdx.x & 31;
    const int tilesN = NOUT / 16;
    const long tile = (long)blockIdx.x * 8 + wave;
    const int tm = (int)(tile / tilesN);
    const int tn = (int)(tile % tilesN);
    if (tm * 16 >= M) return;

    const int half = lane >> 4;   // selects K pair within each K-step of 4
    const int l15  = lane & 15;
    const int arow = tm * 16 + l15;
    const int col  = tn * 16 + l15;

    const float* Ap = A + (long)arow * K + half * 2;      // + k0 each iter
    const float* Wp = W + (long)(half * 2) * NOUT + col;  // + k0*NOUT each iter

    v8f acc = {0.f, 0.f, 0.f, 0.f, 0.f, 0.f, 0.f, 0.f};
#pragma unroll
    for (int k0 = 0; k0 < K; k0 += 4) {
        v2f a = *(const v2f*)(Ap + k0);          // A[arow][k0+2h .. +1]
        v2f b;
        b.x = Wp[(long)k0 * NOUT];               // W[k0+2h  ][col]
        b.y = Wp[(long)(k0 + 1) * NOUT];         // W[k0+2h+1][col]
        acc = __builtin_amdgcn_wmma_f32_16x16x4_f32(
            /*neg_a=*/false, a, /*neg_b=*/false, b,
            /*c_mod=*/(short)0, acc, /*reuse_a=*/false, /*reuse_b=*/false);
    }

    float* Cp = C + (long)(tm * 16 + half * 8) * NOUT + col;
#pragma unroll
    for (int g = 0; g < 8; ++g)
        Cp[(long)g * NOUT] = acc[g];
}

// ---------------------------------------------------------------------------
// Edge aggregation: agg[dst] += dinv[src]*dinv[dst] * h[src]
// One edge per C/4 consecutive threads; float4 gather + 4 f32 atomics.
// h (102MB / 51MB) fits in the 192MB L2, so gathers/atomics run at L2 speed.
// ---------------------------------------------------------------------------
template<int C>
__global__ __launch_bounds__(256)
void aggregate_kernel(const float* __restrict__ h, const float* __restrict__ dinv,
                      const int* __restrict__ src, const int* __restrict__ dst,
                      float* __restrict__ agg, int E) {
    constexpr int CH = C / 4;
    long tid = (long)blockIdx.x * blockDim.x + threadIdx.x;
    long e   = tid / CH;
    int  c4  = (int)(tid % CH);
    if (e >= E) return;
    const int s = src[e];
    const int d = dst[e];
    const float nrm = dinv[s] * dinv[d];
    const float4 v = *(const float4*)(h + (long)s * C + c4 * 4);
    float* p = agg + (long)d * C + c4 * 4;
    unsafeAtomicAdd(p + 0, v.x * nrm);
    unsafeAtomicAdd(p + 1, v.y * nrm);
    unsafeAtomicAdd(p + 2, v.z * nrm);
    unsafeAtomicAdd(p + 3, v.w * nrm);
}

// ---------------------------------------------------------------------------
// Epilogue: agg[i] (+)= h[i]*dinv[row]^2 + b[col]; optional ReLU; in place.
// ---------------------------------------------------------------------------
template<int C, bool RELU>
__global__ __launch_bounds__(256)
void self_bias_kernel(float* __restrict__ agg, const float* __restrict__ h,
                      const float* __restrict__ dinv, const float* __restrict__ b,
                      int Nn) {
    long i = (long)blockIdx.x * blockDim.x + threadIdx.x;
    if (i >= (long)Nn * C) return;
    const int row = (int)(i / C);
    const int col = (int)(i % C);
    const float di = dinv[row];
    float v = agg[i] + h[i] * (di * di) + b[col];
    agg[i] = RELU ? fmaxf(v, 0.0f) : v;
}

// ---------------------------------------------------------------------------
extern "C" void kernel_launch(void* const* d_in, const int* in_sizes, int n_in,
                              void* d_out, int out_size, void* d_ws, size_t ws_size,
                              hipStream_t stream) {
    constexpr int INC  = 128;
    constexpr int HIDC = 256;
    constexpr int OUTC = 128;

    const float* x  = (const float*)d_in[0];
    const int*   ei = (const int*)  d_in[1];
    const float* W1 = (const float*)d_in[2];
    const float* b1 = (const float*)d_in[3];
    const float* W2 = (const float*)d_in[4];
    const float* b2 = (const float*)d_in[5];

    const int Nn = in_sizes[0] / INC;   // 100000
    const int E  = in_sizes[1] / 2;     // 1600000
    const int* src = ei;
    const int* dst = ei + E;
    float* out = (float*)d_out;

    // Workspace layout
    float* dinv = (float*)d_ws;                       // N      (deg, then dinv)
    float* h1   = dinv + Nn;                          // N*256
    float* agg1 = h1 + (long)Nn * HIDC;               // N*256  (becomes layer-2 input)
    float* h2   = agg1 + (long)Nn * HIDC;             // N*128

    // Zero accumulators (graph-capturable async memsets)
    hipMemsetAsync(dinv, 0, (size_t)Nn * sizeof(float), stream);
    hipMemsetAsync(agg1, 0, (size_t)Nn * HIDC * sizeof(float), stream);
    hipMemsetAsync(out,  0, (size_t)out_size * sizeof(float), stream);

    // Degrees -> dinv
    degree_kernel<<<(E + 255) / 256, 256, 0, stream>>>(dst, dinv, E);
    rsqrt_kernel<<<(Nn + 255) / 256, 256, 0, stream>>>(dinv, Nn);

    // ---- Layer 1 ----
    {
        long tiles  = (long)((Nn + 15) / 16) * (HIDC / 16);
        long blocks = (tiles + 7) / 8;
        gemm_wmma_f32<INC, HIDC><<<(int)blocks, 256, 0, stream>>>(x, W1, h1, Nn);
    }
    {
        long threads = (long)E * (HIDC / 4);
        aggregate_kernel<HIDC><<<(int)((threads + 255) / 256), 256, 0, stream>>>(
            h1, dinv, src, dst, agg1, E);
    }
    {
        long n = (long)Nn * HIDC;
        self_bias_kernel<HIDC, true><<<(int)((n + 255) / 256), 256, 0, stream>>>(
            agg1, h1, dinv, b1, Nn);
    }

    // ---- Layer 2 (input = agg1) ----
    {
        long tiles  = (long)((Nn + 15) / 16) * (OUTC / 16);
        long blocks = (tiles + 7) / 8;
        gemm_wmma_f32<HIDC, OUTC><<<(int)blocks, 256, 0, stream>>>(agg1, W2, h2, Nn);
    }
    {
        long threads = (long)E * (OUTC / 4);
        aggregate_kernel<OUTC><<<(int)((threads + 255) / 256), 256, 0, stream>>>(
            h2, dinv, src, dst, out, E);
    }
    {
        long n = (long)Nn * OUTC;
        self_bias_kernel<OUTC, false><<<(int)((n + 255) / 256), 256, 0, stream>>>(
            out, h2, dinv, b2, Nn);
    }
}